// RPTNv2_72962904425171
// MI455X (gfx1250) — compile-verified
//
#include <hip/hip_runtime.h>
#include <stdint.h>

#define IN_CH   16
#define OUT_CH  64
#define EXP_CH  4096
#define SET_LEN 64
#define NB      8
#define NH      56
#define NW      56
#define NPIX        (NH * NW)          // 3136
#define NPIX_TILES  (NPIX / 16)        // 196
#define PIX_CHUNKS  4
#define TILES_PER_WAVE (NPIX_TILES / PIX_CHUNKS) // 49
#define CHUNK_PIX   (TILES_PER_WAVE * 16)        // 784 = 14 rows
#define N_OUT   (NB * OUT_CH * NPIX)   // 1605632
#define LDS_W   58                      // 56 + 2 zero pad cols
#define LDS_H   16                      // 14 rows + 2 halo rows
#define LDS_N   (LDS_H * LDS_W)         // 928

typedef __attribute__((ext_vector_type(2))) float v2f;
typedef __attribute__((ext_vector_type(8))) float v8f;

// Order-preserving float -> u32 transform so atomicMax(u32) == float max.
__device__ __forceinline__ unsigned xform(float f) {
    unsigned u = __float_as_uint(f);
    return (u & 0x80000000u) ? ~u : (u | 0x80000000u);
}

__global__ void k_init(unsigned* out_t, int* inv_count, int* inv_cursor) {
    int i = blockIdx.x * blockDim.x + threadIdx.x;
    if (i < N_OUT) out_t[i] = 0u;              // transformed identity for max
    if (i < EXP_CH) { inv_count[i] = 0; inv_cursor[i] = 0; }
}

__global__ void k_count(const int* __restrict__ idx, int* inv_count) {
    int i = blockIdx.x * blockDim.x + threadIdx.x;
    if (i < OUT_CH * SET_LEN) {
        int e = idx[i] & (EXP_CH - 1);
        atomicAdd(&inv_count[e], 1);
    }
}

__global__ void k_scan(const int* __restrict__ inv_count, int* inv_off) {
    if (blockIdx.x == 0 && threadIdx.x == 0) {
        int acc = 0;
        for (int e = 0; e < EXP_CH; ++e) { inv_off[e] = acc; acc += inv_count[e]; }
        inv_off[EXP_CH] = acc;
    }
}

__global__ void k_fill(const int* __restrict__ idx, const int* __restrict__ inv_off,
                       int* inv_cursor, int* inv_list) {
    int i = blockIdx.x * blockDim.x + threadIdx.x;
    if (i < OUT_CH * SET_LEN) {
        int e = idx[i] & (EXP_CH - 1);
        int pos = inv_off[e] + atomicAdd(&inv_cursor[e], 1);
        inv_list[pos] = i / SET_LEN;           // output channel oc
    }
}

// One block = (b, c, pixel-chunk). 16 waves cover the 16 e-tiles of channel c.
__global__ void __launch_bounds__(512)
k_conv_scatter(const float* __restrict__ x, const float* __restrict__ weight,
               const float* __restrict__ bias, const int* __restrict__ inv_off,
               const int* __restrict__ inv_list, unsigned* __restrict__ out_t) {
    __shared__ float tile[LDS_N];

    const int bid = blockIdx.x;
    const int q = bid & (PIX_CHUNKS - 1);      // pixel chunk (14 image rows)
    const int c = (bid >> 2) & (IN_CH - 1);    // input channel
    const int b = bid >> 6;                    // batch

    const float* xp = x + ((size_t)b * IN_CH + c) * NPIX;

    // ---- stage rows [14q-1 .. 14q+14] of x-plane into LDS, zero-padded on all sides ----
    const int h0 = q * 14 - 1;
    for (int i = threadIdx.x; i < LDS_N; i += 512) {
        int r = i / LDS_W, col = i - r * LDS_W;
        int y = h0 + r, xx = col - 1;
        bool ok = ((unsigned)y < (unsigned)NH) && ((unsigned)xx < (unsigned)NW);
        int yc = min(max(y, 0), NH - 1);
        int xc = min(max(xx, 0), NW - 1);
        float v = xp[yc * NW + xc];
        tile[i] = ok ? v : 0.0f;
    }
    __syncthreads();

    const int et   = threadIdx.x >> 5;         // e-tile 0..15
    const int lane = threadIdx.x & 31;
    const int hi   = lane >> 4;                // 16-lane half
    const int m    = lane & 15;

    const int e_base = c * 256 + et * 16;

    // A matrix: 16 filters x K(=9 padded to 12). lanes 0-15: K={0,1}, lanes 16-31: K={2,3} per chunk.
    const float* wp = weight + (size_t)(e_base + m) * 9;
    v2f a0, a1, a2;
    a0.x = wp[2 * hi];     a0.y = wp[2 * hi + 1];
    a1.x = wp[4 + 2 * hi]; a1.y = wp[5 + 2 * hi];
    a2.x = hi ? 0.0f : wp[8];
    a2.y = 0.0f;

    // Hoisted per-D-row metadata (rows M = v + 8*hi): bias + first two CSR targets.
    int      cntv[8], joff[8];
    unsigned oc0n[8], oc1n[8];
    float    bv[8];
#pragma unroll
    for (int v = 0; v < 8; ++v) {
        int e   = e_base + v + 8 * hi;
        int o0  = inv_off[e];
        int cnt = inv_off[e + 1] - o0;
        joff[v] = o0;
        cntv[v] = cnt;
        oc0n[v] = (cnt > 0) ? (unsigned)inv_list[o0] * NPIX : 0u;
        oc1n[v] = (cnt > 1) ? (unsigned)inv_list[o0 + 1] * NPIX : 0u;
        bv[v]   = bias[e];
    }

    // Per-half 3x3 neighbor offsets in the padded LDS tile.
    // hi=0 covers taps k={0,1,4,5,8}; hi=1 covers k={2,3,6,7}.
    const int   d0 = hi ? -57 : -59;   // k2 / k0
    const int   d1 = hi ?  -1 : -58;   // k3 / k1
    const int   d2 = hi ?  57 :   0;   // k6 / k4
    const int   d3 = hi ?  58 :   1;   // k7 / k5
    const float m4 = hi ? 0.0f : 1.0f; //      k8 mask

    const unsigned obase0 = (unsigned)b * (OUT_CH * NPIX) + (unsigned)(q * CHUNK_PIX);

    for (int pt = 0; pt < TILES_PER_WAVE; ++pt) {
        const int pp = pt * 16 + m;                 // pixel within chunk (column N = m)
        const int rr = pp / NW, ww = pp - rr * NW;
        const int lidx = (rr + 1) * LDS_W + ww + 1;

        v2f b0, b1, b2;
        b0.x = tile[lidx + d0];
        b0.y = tile[lidx + d1];
        b1.x = tile[lidx + d2];
        b1.y = tile[lidx + d3];
        b2.x = tile[lidx + 59] * m4;
        b2.y = 0.0f;

        v8f acc = {0.f, 0.f, 0.f, 0.f, 0.f, 0.f, 0.f, 0.f};
        acc = __builtin_amdgcn_wmma_f32_16x16x4_f32(false, a0, false, b0, (short)0, acc, false, false);
        acc = __builtin_amdgcn_wmma_f32_16x16x4_f32(false, a1, false, b1, (short)0, acc, false, false);
        acc = __builtin_amdgcn_wmma_f32_16x16x4_f32(false, a2, false, b2, (short)0, acc, false, false);

        const unsigned ob = obase0 + (unsigned)pp;  // 32-bit output offset
#pragma unroll
        for (int v = 0; v < 8; ++v) {
            float    val = acc[v] + bv[v];
            unsigned tv  = xform(val);
            if (cntv[v] > 0) atomicMax(&out_t[ob + oc0n[v]], tv);
            if (cntv[v] > 1) atomicMax(&out_t[ob + oc1n[v]], tv);
            if (cntv[v] > 2) {                      // rare tail (Poisson λ=1)
                int j1 = joff[v] + cntv[v];
                for (int j = joff[v] + 2; j < j1; ++j)
                    atomicMax(&out_t[ob + (unsigned)inv_list[j] * NPIX], tv);
            }
        }
    }
}

__global__ void k_fin(const unsigned* __restrict__ out_t, float* __restrict__ out) {
    int i = blockIdx.x * blockDim.x + threadIdx.x;
    if (i < N_OUT) {
        unsigned tv = out_t[i];
        unsigned u  = (tv & 0x80000000u) ? (tv & 0x7FFFFFFFu) : ~tv;
        out[i] = __uint_as_float(u);
    }
}

extern "C" void kernel_launch(void* const* d_in, const int* in_sizes, int n_in,
                              void* d_out, int out_size, void* d_ws, size_t ws_size,
                              hipStream_t stream) {
    (void)in_sizes; (void)n_in; (void)out_size; (void)ws_size;

    const float* x      = (const float*)d_in[0];
    const float* weight = (const float*)d_in[1];
    const float* bias   = (const float*)d_in[2];
    const int*   idx    = (const int*)d_in[3];
    float*       out    = (float*)d_out;

    unsigned* out_t     = (unsigned*)d_ws;                 // N_OUT u32
    int* inv_count      = (int*)(out_t + N_OUT);           // 4096
    int* inv_off        = inv_count + EXP_CH;              // 4097
    int* inv_cursor     = inv_off + (EXP_CH + 1);          // 4096
    int* inv_list       = inv_cursor + EXP_CH;             // 4096

    k_init <<<(N_OUT + 255) / 256, 256, 0, stream>>>(out_t, inv_count, inv_cursor);
    k_count<<<(OUT_CH * SET_LEN + 255) / 256, 256, 0, stream>>>(idx, inv_count);
    k_scan <<<1, 1, 0, stream>>>(inv_count, inv_off);
    k_fill <<<(OUT_CH * SET_LEN + 255) / 256, 256, 0, stream>>>(idx, inv_off, inv_cursor, inv_list);

    // One block per (b, c, pixel-chunk): 8*16*4 = 512 blocks x 16 waves.
    k_conv_scatter<<<NB * IN_CH * PIX_CHUNKS, 512, 0, stream>>>(x, weight, bias,
                                                                inv_off, inv_list, out_t);

    k_fin<<<(N_OUT + 255) / 256, 256, 0, stream>>>(out_t, out);
}